// ContrastiveLoss_51943334478253
// MI455X (gfx1250) — compile-verified
//
#include <hip/hip_runtime.h>
#include <math.h>

typedef float v2f __attribute__((ext_vector_type(2)));
typedef float v8f __attribute__((ext_vector_type(8)));

#define BATCH 4
#define NROW  1000
#define NPAD  1024          // padded rows (zero-filled 1000..1023)
#define CH    2048
#define NTILE 63            // 63 row-tiles of 16 cover rows 0..1007
#define INV_TEMP (1.0f/0.07f)

// LDS layout in float units
#define A_STRIDE 2052                     // 2048 + 4 pad -> conflict-free ds_load_b64
#define A_OFF    0
#define SIM_OFF  (16*A_STRIDE)            // 32832
#define TGT_OFF  (SIM_OFF + 16*NPAD)      // 49216 (ints stored here)
#define ROWT_OFF (TGT_OFF + NPAD)         // 50240
#define ROWP_OFF (ROWT_OFF + 16)          // 50256
#define LDS_FLOATS (ROWP_OFF + 16)        // 50272 floats = 201088 bytes

// ---------------- Kernel 1: row L2-normalize into padded workspace ----------------
__global__ void cl_normalize(const float* __restrict__ feat, float* __restrict__ fn) {
    const int row = blockIdx.x;           // 0 .. BATCH*NPAD-1
    const int b = row >> 10;
    const int i = row & (NPAD - 1);
    float* out = fn + (size_t)row * CH;
    if (i >= NROW) {
        for (int c = threadIdx.x; c < CH; c += 256) out[c] = 0.0f;
        return;
    }
    const float* f = feat + ((size_t)(b * NROW + i)) * CH;
    float ss = 0.0f;
    for (int c = threadIdx.x; c < CH; c += 256) { float v = f[c]; ss += v * v; }
    __shared__ float red[8];
    #pragma unroll
    for (int o = 16; o; o >>= 1) ss += __shfl_xor(ss, o, 32);
    const int wave = threadIdx.x >> 5, lane = threadIdx.x & 31;
    if (lane == 0) red[wave] = ss;
    __syncthreads();
    float tot = 0.0f;
    #pragma unroll
    for (int w = 0; w < 8; ++w) tot += red[w];
    float nrm = fmaxf(sqrtf(tot), 1e-12f);
    float inv = 1.0f / nrm;
    for (int c = threadIdx.x; c < CH; c += 256) out[c] = f[c] * inv;
}

// ---------------- Kernel 2: fused WMMA-f32 GEMM strip + loss reduction ----------------
__global__ void cl_gemm_reduce(const float* __restrict__ fn,
                               const int* __restrict__ tgt,
                               float* __restrict__ partials) {
    extern __shared__ float lds[];
    const int mi   = blockIdx.x;          // row tile 0..62
    const int b    = blockIdx.y;          // sample 0..3
    const int tid  = threadIdx.x;
    const int lane = tid & 31;
    const int wave = tid >> 5;
    const int mrow = lane & 15;           // M (or N) within tile
    const int half = lane >> 4;           // selects K pair per ISA 32-bit A/B layout

    // Stage targets for this sample into LDS
    int* tl = (int*)&lds[TGT_OFF];
    for (int j = tid; j < NPAD; j += 256) tl[j] = (j < NROW) ? tgt[b * NROW + j] : -1;

    // Stage A strip (16 rows x 2048) into LDS with padded stride
    {
        const float* src = fn + ((size_t)(b * NPAD + mi * 16)) * CH;
        for (int idx = tid; idx < (16 * CH) / 4; idx += 256) {
            int flat = idx * 4;
            int row = flat >> 11;         // /2048
            int col = flat & (CH - 1);
            float4 v = *(const float4*)(src + (size_t)row * CH + col);
            *(float4*)&lds[A_OFF + row * A_STRIDE + col] = v;
        }
    }
    __syncthreads();

    // GEMM: each wave computes 8 column tiles of 16 (total 64 tiles = 1024 cols)
    const float* Ap = &lds[A_OFF + mrow * A_STRIDE + 2 * half];
    for (int p = 0; p < 8; ++p) {
        const int nj = p * 8 + wave;      // 0..63
        const float* Bp = fn + ((size_t)(b * NPAD + nj * 16 + mrow)) * CH + 2 * half;
        v8f acc = {};
        #pragma unroll 8
        for (int k0 = 0; k0 < CH; k0 += 4) {
            v2f av = *(const v2f*)(Ap + k0);        // ds_load_b64
            v2f bv = *(const v2f*)(Bp + k0);        // global_load_b64
            acc = __builtin_amdgcn_wmma_f32_16x16x4_f32(
                false, av, false, bv, (short)0, acc, false, false);
        }
        // C/D layout: VGPR r -> row r (lanes 0-15) / row r+8 (lanes 16-31), col = mrow
        float* so = &lds[SIM_OFF + (8 * half) * NPAD + nj * 16 + mrow];
        #pragma unroll
        for (int r = 0; r < 8; ++r) so[(size_t)r * NPAD] = acc[r];
    }
    __syncthreads();

    // Loss reduction: each wave owns 2 rows of the strip
    for (int rr = 0; rr < 2; ++rr) {
        const int r = wave * 2 + rr;
        const int i = mi * 16 + r;        // global row index
        float T = 0.0f, P = 0.0f;
        if (i < NROW) {
            const int ti = tl[i];
            const float* srow = &lds[SIM_OFF + r * NPAD];
            // pass 1: S_i = sum over negatives of exp(sim)
            float sacc = 0.0f;
            for (int j = lane; j < NROW; j += 32) {
                float s = srow[j] * INV_TEMP;
                sacc += (tl[j] != ti) ? expf(s) : 0.0f;
            }
            #pragma unroll
            for (int o = 16; o; o >>= 1) sacc += __shfl_xor(sacc, o, 32);
            const float S = sacc;
            // pass 2: positive-pair terms
            float tp = 0.0f, pc = 0.0f;
            for (int j = lane; j < NROW; j += 32) {
                if (tl[j] == ti && j != i) {
                    float s = srow[j] * INV_TEMP;
                    tp += logf(expf(s) + S) - s;
                    pc += 1.0f;
                }
            }
            #pragma unroll
            for (int o = 16; o; o >>= 1) {
                tp += __shfl_xor(tp, o, 32);
                pc += __shfl_xor(pc, o, 32);
            }
            T = tp + ((float)NROW - pc) * log1pf(S);
            P = pc;
        }
        if (lane == 0) { lds[ROWT_OFF + r] = T; lds[ROWP_OFF + r] = P; }
    }
    __syncthreads();
    if (tid == 0) {
        float Ts = 0.0f, Ps = 0.0f;
        #pragma unroll
        for (int r = 0; r < 16; ++r) { Ts += lds[ROWT_OFF + r]; Ps += lds[ROWP_OFF + r]; }
        float* pp = partials + ((size_t)b * NTILE + mi) * 2;
        pp[0] = Ts;
        pp[1] = Ps;
    }
}

// ---------------- Kernel 3: final deterministic reduce ----------------
__global__ void cl_final(const float* __restrict__ partials, float* __restrict__ out) {
    if (threadIdx.x == 0 && blockIdx.x == 0) {
        float total = 0.0f;
        int nv = 0;
        for (int b = 0; b < BATCH; ++b) {
            float T = 0.0f, P = 0.0f;
            for (int t = 0; t < NTILE; ++t) {
                T += partials[((size_t)b * NTILE + t) * 2 + 0];
                P += partials[((size_t)b * NTILE + t) * 2 + 1];
            }
            if (P > 0.0f) { total += T / (P + 1e-6f); ++nv; }
        }
        out[0] = (nv > 0) ? (0.1f * total / (float)nv) : (0.1f * 0.1f);
    }
}

extern "C" void kernel_launch(void* const* d_in, const int* in_sizes, int n_in,
                              void* d_out, int out_size, void* d_ws, size_t ws_size,
                              hipStream_t stream) {
    (void)in_sizes; (void)n_in; (void)out_size; (void)ws_size;
    const float* features = (const float*)d_in[0];
    const int*   target   = (const int*)d_in[1];

    float* fn       = (float*)d_ws;                       // [4][1024][2048] f32 = 33.5 MB
    float* partials = fn + (size_t)BATCH * NPAD * CH;     // [4][63][2] f32

    cl_normalize<<<dim3(BATCH * NPAD), 256, 0, stream>>>(features, fn);

    const size_t ldsBytes = (size_t)LDS_FLOATS * sizeof(float);   // 201088 B < 320 KB
    cl_gemm_reduce<<<dim3(NTILE, BATCH), 256, ldsBytes, stream>>>(fn, target, partials);

    cl_final<<<1, 32, 0, stream>>>(partials, (float*)d_out);
}